// SoftmaxPooling_57827439673421
// MI455X (gfx1250) — compile-verified
//
#include <hip/hip_runtime.h>

typedef __attribute__((ext_vector_type(2))) float v2f;
typedef __attribute__((ext_vector_type(8))) float v8f;

#define DIM 128          // feature dimension (fixed by reference)
#define CHUNK 128        // nodes per block

// D(16x16) = A(16x4) * B(4x16) + C, all fp32. wave32.
__device__ __forceinline__ v8f wmma_f32_4(v2f a, v2f b, v8f c) {
    return __builtin_amdgcn_wmma_f32_16x16x4_f32(
        /*neg_a=*/false, a, /*neg_b=*/false, b,
        /*c_mod=*/(short)0, c, /*reuse_a=*/false, /*reuse_b=*/false);
}

__global__ __launch_bounds__(128)
void segsum_wmma_kernel(const float* __restrict__ X,
                        const float* __restrict__ Xatt,
                        const int*   __restrict__ gidx,
                        float* __restrict__ ysum,   // [n_graphs, DIM]  (d_out)
                        float* __restrict__ asum,   // [n_graphs, DIM]  (d_ws)
                        int n_nodes, int n_graphs)
{
    const int lane  = threadIdx.x & 31;
    const int wv    = threadIdx.x >> 5;      // 4 waves, each owns 32 columns
    const int cbase = wv * 32;
    const int half  = lane >> 4;             // 0: lanes 0-15 (K=0,1), 1: lanes 16-31 (K=2,3)
    const int m     = lane & 15;             // N index in B / M index in A

    const int start = blockIdx.x * CHUNK;
    if (start >= n_nodes) return;
    const int end = min(start + CHUNK, n_nodes);

    v8f acc_a0 = {}, acc_a1 = {}, acc_w0 = {}, acc_w1 = {};
    int win = __builtin_amdgcn_readfirstlane(gidx[start]);   // segment id of window slot 0

    // flush register window [win, win+16) with one atomic per nonzero value
    auto flush = [&]() {
        const int c0 = cbase + m;
#pragma unroll
        for (int j = 0; j < 8; ++j) {
            const int seg = win + j + half * 8;    // C/D layout: vgpr j -> M=j (lo) / j+8 (hi)
            if (seg < n_graphs) {
                float va0 = acc_a0[j], vw0 = acc_w0[j];
                float va1 = acc_a1[j], vw1 = acc_w1[j];
                if (va0 != 0.0f) atomicAdd(&asum[seg * DIM + c0],      va0);
                if (vw0 != 0.0f) atomicAdd(&ysum[seg * DIM + c0],      vw0);
                if (va1 != 0.0f) atomicAdd(&asum[seg * DIM + c0 + 16], va1);
                if (vw1 != 0.0f) atomicAdd(&ysum[seg * DIM + c0 + 16], vw1);
            }
        }
        v8f z = {};
        acc_a0 = z; acc_a1 = z; acc_w0 = z; acc_w1 = z;
    };

    // scalar fallback: direct atomic accumulation for cnt nodes starting at n0
    auto fallback = [&](int n0, int cnt) {
        const int c = cbase + lane;              // 32 columns per wave
        for (int k = 0; k < cnt; ++k) {
            const int g = __builtin_amdgcn_readfirstlane(gidx[n0 + k]);
            const size_t row = (size_t)(n0 + k) * DIM;
            float a = __expf(Xatt[row + c]);
            float x = X[row + c];
            atomicAdd(&asum[g * DIM + c], a);
            atomicAdd(&ysum[g * DIM + c], a * x);
        }
    };

    for (int n0 = start; n0 < end; n0 += 4) {
        const int rem = end - n0;
        if (rem < 4) { fallback(n0, rem); break; }

        const int g0 = __builtin_amdgcn_readfirstlane(gidx[n0 + 0]);
        const int g1 = __builtin_amdgcn_readfirstlane(gidx[n0 + 1]);
        const int g2 = __builtin_amdgcn_readfirstlane(gidx[n0 + 2]);
        const int g3 = __builtin_amdgcn_readfirstlane(gidx[n0 + 3]);

        if (g3 - win >= 16) {                    // window full -> flush & restart at g0
            flush();
            win = g0;
            if (g3 - g0 >= 16) {                 // pathological: >16 graphs in 4 nodes
                fallback(n0, 4);
                continue;
            }
        }

        // Load B operands: lane supplies rows (n0+2*half, n0+2*half+1), cols cbase+m and +16.
        const size_t r0 = (size_t)(n0 + half * 2) * DIM;
        const size_t r1 = r0 + DIM;
        const int c0 = cbase + m, c1 = c0 + 16;

        if (n0 + 8 < end) {                      // stream-ahead prefetch (global_prefetch_b8)
            __builtin_prefetch(&X[(size_t)(n0 + 8) * DIM + cbase + lane], 0, 1);
            __builtin_prefetch(&Xatt[(size_t)(n0 + 8) * DIM + cbase + lane], 0, 1);
        }

        float e00 = __expf(Xatt[r0 + c0]);
        float e10 = __expf(Xatt[r1 + c0]);
        float e01 = __expf(Xatt[r0 + c1]);
        float e11 = __expf(Xatt[r1 + c1]);
        float x00 = X[r0 + c0], x10 = X[r1 + c0];
        float x01 = X[r0 + c1], x11 = X[r1 + c1];

        v2f ba0 = {e00, e10};
        v2f ba1 = {e01, e11};
        v2f bw0 = {e00 * x00, e10 * x10};
        v2f bw1 = {e01 * x01, e11 * x11};

        // One-hot selector A: row m gets node k iff its graph lands in window slot m.
        const int glo = half ? g2 : g0;
        const int ghi = half ? g3 : g1;
        v2f a = { (glo - win == m) ? 1.0f : 0.0f,
                  (ghi - win == m) ? 1.0f : 0.0f };

        acc_a0 = wmma_f32_4(a, ba0, acc_a0);
        acc_a1 = wmma_f32_4(a, ba1, acc_a1);
        acc_w0 = wmma_f32_4(a, bw0, acc_w0);
        acc_w1 = wmma_f32_4(a, bw1, acc_w1);
    }

    flush();
}

__global__ void finalize_kernel(float* __restrict__ y,
                                const float* __restrict__ asum, int n)
{
    for (int i = blockIdx.x * blockDim.x + threadIdx.x; i < n;
         i += gridDim.x * blockDim.x)
        y[i] = y[i] / asum[i];               // 0/0 -> NaN matches reference for empty graphs
}

extern "C" void kernel_launch(void* const* d_in, const int* in_sizes, int n_in,
                              void* d_out, int out_size, void* d_ws, size_t ws_size,
                              hipStream_t stream) {
    const float* X    = (const float*)d_in[0];
    const float* Xatt = (const float*)d_in[1];
    const int*   gidx = (const int*)d_in[2];
    const int n_nodes  = in_sizes[2];
    const int n_graphs = in_sizes[3];

    float* ysum = (float*)d_out;             // y segment-sum accumulates here
    float* asum = (float*)d_ws;              // att segment-sum in scratch

    hipMemsetAsync(ysum, 0, (size_t)out_size * sizeof(float), stream);
    hipMemsetAsync(asum, 0, (size_t)n_graphs * DIM * sizeof(float), stream);

    const int blocks = (n_nodes + CHUNK - 1) / CHUNK;
    segsum_wmma_kernel<<<blocks, 128, 0, stream>>>(X, Xatt, gidx, ysum, asum,
                                                   n_nodes, n_graphs);

    const int fin_threads = 256;
    const int fin_blocks  = (out_size + fin_threads - 1) / fin_threads;
    finalize_kernel<<<fin_blocks, fin_threads, 0, stream>>>(ysum, asum, out_size);
}